// PINOModel_53188874993842
// MI455X (gfx1250) — compile-verified
//
#include <hip/hip_runtime.h>
#include <math.h>

typedef __attribute__((ext_vector_type(2))) float v2f;
typedef __attribute__((ext_vector_type(8))) float v8f;

__device__ __forceinline__ float gelu_exact(float x) {
  return 0.5f * x * (1.0f + erff(x * 0.7071067811865476f));
}

// ---------------------------------------------------------------------------
// Generic batched dual-product WMMA GEMM:
//   C[b] = A0[b] @ B0[b] + sgn1 * (A1[b] @ B1[b])   (+ bias) (+ exact GELU)
// A: (M x K) row-major lda, B: (K x N) row-major ldb, C: (M x N) row-major ldc.
// One wave32 per 16x16 C tile; K-loop of V_WMMA_F32_16X16X4_F32.
// A tiles are staged zero-padded in LDS (sgn1 baked into the A1 tile), so the
// inner loop is branch-free: ds_load (A) + coalesced global_load (B) + wmma.
// Out-of-range N columns are computed on clamped addresses but never stored
// (column j of B only ever feeds column j of D).
// Requires K <= 128 (all launches below satisfy this).
// ---------------------------------------------------------------------------
__global__ __launch_bounds__(32)
void wmma_gemm_dual(const float* __restrict__ A0, const float* __restrict__ B0,
                    const float* __restrict__ A1, const float* __restrict__ B1,
                    const float* __restrict__ bias, float* __restrict__ C,
                    int M, int N, int K, int lda, int ldb, int ldc,
                    long long bsA, long long bsB, long long bsC,
                    float sgn1, int biasMode, int actGelu)
{
  __shared__ float sA[2 * 16 * 128];   // [0]: A0 tile, [2048]: sgn1*A1 tile

  const int tilesM = (M + 15) >> 4;
  const int tm = (int)(blockIdx.x % (unsigned)tilesM) << 4;
  const int tn = (int)(blockIdx.x / (unsigned)tilesM) << 4;
  const long long zb = blockIdx.z;
  const float* a0 = A0 + zb * bsA;
  const float* b0 = B0 + zb * bsB;
  float* c = C + zb * bsC;

  const int lane = threadIdx.x;
  const int half = lane >> 4;     // 0: K pair {0,1}; 1: K pair {2,3}
  const int l16  = lane & 15;
  const int K4   = (K + 3) & ~3;  // zero-padded K

  // ---- stage A tiles into LDS (branch-free: clamped loads + select) ----
  {
    const int nChains = (A1 != nullptr) ? 2 : 1;
    for (int ch = 0; ch < nChains; ++ch) {
      const float* ap = (ch == 0) ? a0 : (A1 + zb * bsA);
      const float  sg = (ch == 0) ? 1.f : sgn1;
      float* dst = sA + ch * 2048;
      for (int r = 0; r < 16; ++r) {
        const int  rr  = tm + r;
        const int  rc  = (rr < M) ? rr : (M - 1);
        const bool rok = (rr < M);
        const float* arow = ap + (long long)rc * lda;
        for (int cidx = lane; cidx < K4; cidx += 32) {
          const int cc = (cidx < K) ? cidx : (K - 1);
          float v = arow[cc];                       // always in-bounds
          dst[r * K4 + cidx] = (rok && cidx < K) ? (sg * v) : 0.f;
        }
      }
    }
  }

  // ---- B column pointers (clamped; OOB columns never stored) ----
  const int cb  = tn + l16;
  const int cbc = (cb < N) ? cb : (N - 1);
  const long long ldbl = ldb;

  v8f acc = {};

  // ---- chain 0: main loop (no bounds checks) + clamped tail ----
  {
    const float* sp = sA + l16 * K4 + (half << 1);
    const float* bp = b0 + cbc + (long long)(half << 1) * ldbl;
    int k = 0;
    for (; k + 4 <= K; k += 4) {
      v2f av, bv;
      av.x = sp[0];
      av.y = sp[1];
      bv.x = bp[0];
      bv.y = bp[ldbl];
      __builtin_prefetch(bp + 8 * ldbl, 0, 1);      // global_prefetch_b8
      acc = __builtin_amdgcn_wmma_f32_16x16x4_f32(false, av, false, bv,
                                                  (short)0, acc, false, false);
      sp += 4;
      bp += 4 * ldbl;
    }
    if (k < K4) {                                   // tail: A is zero-padded
      const int ka = k + (half << 1);
      const int k0 = (ka     < K) ? ka     : (K - 1);
      const int k1 = (ka + 1 < K) ? ka + 1 : (K - 1);
      v2f av, bv;
      av.x = sp[0];
      av.y = sp[1];
      bv.x = b0[(long long)k0 * ldbl + cbc];
      bv.y = b0[(long long)k1 * ldbl + cbc];
      acc = __builtin_amdgcn_wmma_f32_16x16x4_f32(false, av, false, bv,
                                                  (short)0, acc, false, false);
    }
  }

  // ---- chain 1 (optional): sgn1 already baked into staged A1 ----
  if (A1 != nullptr) {
    const float* b1 = B1 + zb * bsB;
    const float* sp = sA + 2048 + l16 * K4 + (half << 1);
    const float* bp = b1 + cbc + (long long)(half << 1) * ldbl;
    int k = 0;
    for (; k + 4 <= K; k += 4) {
      v2f av, bv;
      av.x = sp[0];
      av.y = sp[1];
      bv.x = bp[0];
      bv.y = bp[ldbl];
      acc = __builtin_amdgcn_wmma_f32_16x16x4_f32(false, av, false, bv,
                                                  (short)0, acc, false, false);
      sp += 4;
      bp += 4 * ldbl;
    }
    if (k < K4) {
      const int ka = k + (half << 1);
      const int k0 = (ka     < K) ? ka     : (K - 1);
      const int k1 = (ka + 1 < K) ? ka + 1 : (K - 1);
      v2f av, bv;
      av.x = sp[0];
      av.y = sp[1];
      bv.x = b1[(long long)k0 * ldbl + cbc];
      bv.y = b1[(long long)k1 * ldbl + cbc];
      acc = __builtin_amdgcn_wmma_f32_16x16x4_f32(false, av, false, bv,
                                                  (short)0, acc, false, false);
    }
  }

  // ---- epilogue: bias + GELU + bounds-checked store ----
#pragma unroll
  for (int r = 0; r < 8; ++r) {
    const int row = tm + r + (half << 3);
    const int col = tn + l16;
    if (row < M && col < N) {
      float v = acc[r];
      if (biasMode == 1)      v += bias[row];
      else if (biasMode == 2) v += bias[col];
      if (actGelu) v = gelu_exact(v);
      c[(long long)row * ldc + col] = v;
    }
  }
}

// DFT matrix generator: re[r*C+c] = coef*cos(th), im[r*C+c] = coef*sin(th)
// th = sign * 2*pi * (r*c mod period) / period ; coef = norm * (dbl && r>0 ? 2 : 1)
__global__ void fill_dft(float* __restrict__ re, float* __restrict__ im,
                         int R, int Cc, int period, float sign, float norm, int dbl)
{
  int idx = blockIdx.x * blockDim.x + threadIdx.x;
  if (idx >= R * Cc) return;
  int r = idx / Cc, c = idx % Cc;
  float coef = norm * ((dbl && r > 0) ? 2.f : 1.f);
  int rc = (int)(((long long)r * c) % period);
  float th = sign * 6.283185307179586f * (float)rc / (float)period;
  re[idx] = coef * cosf(th);
  im[idx] = coef * sinf(th);
}

// dst(C x R) = transpose(src(R x C))
__global__ void transpose2d(const float* __restrict__ src, float* __restrict__ dst,
                            int R, int Cc)
{
  int idx = blockIdx.x * blockDim.x + threadIdx.x;
  if (idx >= R * Cc) return;
  int r = idx / Cc, c = idx % Cc;
  dst[c * R + r] = src[idx];
}

// Spectral channel mix: out[o,m] = sum_i fw[i,o,m] * in[i,m]  (complex, planar)
// fw: (64,64,16,16,16); m = kx*256 + ky*16 + kz; one block per mode, thread = o.
__global__ __launch_bounds__(64)
void mix_modes(const float* __restrict__ fw,
               const float* __restrict__ inr, const float* __restrict__ ini,
               float* __restrict__ outr, float* __restrict__ outi)
{
  const int m = blockIdx.x;    // 0..4095
  const int o = threadIdx.x;   // 0..63
  float ar = 0.f, ai = 0.f;
  for (int i = 0; i < 64; ++i) {
    float w = fw[((((long long)i << 6) | o) << 12) + m];
    ar += w * inr[((long long)i << 12) + m];
    ai += w * ini[((long long)i << 12) + m];
  }
  outr[((long long)o << 12) + m] = ar;
  outi[((long long)o << 12) + m] = ai;
}

// ---------------------------------------------------------------------------
// Host orchestration
// ---------------------------------------------------------------------------
static inline void launch_gemm(hipStream_t s,
    const float* A0, const float* B0, const float* A1, const float* B1,
    const float* bias, float* C,
    int M, int N, int K, int lda, int ldb, int ldc,
    long long bsA, long long bsB, long long bsC, int batch,
    float sgn1, int biasMode, int actGelu)
{
  int tilesM = (M + 15) / 16, tilesN = (N + 15) / 16;
  dim3 grid((unsigned)((long long)tilesM * tilesN), 1u, (unsigned)batch);
  wmma_gemm_dual<<<grid, dim3(32), 0, s>>>(A0, B0, A1, B1, bias, C,
                                           M, N, K, lda, ldb, ldc,
                                           bsA, bsB, bsC, sgn1, biasMode, actGelu);
}

// workspace float offsets
static const size_t OFF_DFTZ_RE = 0;        // 128x16 forward z (cos / -sin)
static const size_t OFF_DFTZ_IM = 2048;
static const size_t OFF_DFTX_RE = 4096;     // 16x128 forward x
static const size_t OFF_DFTX_IM = 6144;
static const size_t OFF_IWX_RE  = 8192;     // 128x16 inverse x (e^{+i}/128)
static const size_t OFF_IWX_IM  = 10240;
static const size_t OFF_CZ_RE   = 12288;    // 16x128 irfft z (harmonic-doubled)
static const size_t OFF_CZ_IM   = 14336;
static const size_t OFF_WYT_RE  = 16384;    // 16xY forward y (per layer)
static const size_t OFF_WYT_IM  = 18432;
static const size_t OFF_IWY_RE  = 20480;    // ny2x16 inverse y (per layer)
static const size_t OFF_IWY_IM  = 22528;
static const size_t OFF_WINT    = 24576;    // 64x4
static const size_t OFF_MWT     = 24832;    // 4 x (64x64)
static const size_t HDR         = 65536;
static const size_t OFF_H  = HDR;                         // activations (C,X,Y,Z) <= 2^27
static const size_t OFF_ZR = OFF_H  + (1ull << 27);       // stage-Z out re  <= 2^24
static const size_t OFF_ZI = OFF_ZR + (1ull << 24);
static const size_t OFF_YR = OFF_ZI + (1ull << 24);       // stage-Y out (64,128,16,16)
static const size_t OFF_YI = OFF_YR + 2097152;
static const size_t OFF_XR = OFF_YI + 2097152;            // modes (64,16,16,16)
static const size_t OFF_XI = OFF_XR + 262144;
static const size_t OFF_ER = OFF_XI + 262144;             // mixed modes
static const size_t OFF_EI = OFF_ER + 262144;
static const size_t OFF_GR = OFF_EI + 262144;             // inv-y out (64,16,ny2,16)
static const size_t OFF_GI = OFF_GR + 1114112;
static const size_t OFF_HR = OFF_GI + 1114112;            // inv-x out (64,128,ny2,16)
static const size_t OFF_HI = OFF_HR + 8650752;
static const size_t OFF_M0 = OFF_HI + 8650752;            // MLP ping-pong (64 x 147456)
static const size_t OFF_M1 = OFF_M0 + 9437184;

extern "C" void kernel_launch(void* const* d_in, const int* in_sizes, int n_in,
                              void* d_out, int out_size, void* d_ws, size_t ws_size,
                              hipStream_t stream)
{
  (void)in_sizes; (void)n_in; (void)out_size; (void)ws_size;
  const float* x     = (const float*)d_in[0];   // (4,128,128,128)
  const float* w_in  = (const float*)d_in[1];   // (4,64)
  const float* b_in  = (const float*)d_in[2];   // (64)
  const float* fws[4] = {(const float*)d_in[3], (const float*)d_in[4],
                         (const float*)d_in[5], (const float*)d_in[6]};
  const float* mw[4] = {(const float*)d_in[7], (const float*)d_in[9],
                        (const float*)d_in[11], (const float*)d_in[13]};
  const float* mb[4] = {(const float*)d_in[8], (const float*)d_in[10],
                        (const float*)d_in[12], (const float*)d_in[14]};
  const float* w_out = (const float*)d_in[15];  // (64,1)
  const float* b_out = (const float*)d_in[16];  // (1)
  float* out = (float*)d_out;
  float* ws  = (float*)d_ws;

  float* dftz_re = ws + OFF_DFTZ_RE; float* dftz_im = ws + OFF_DFTZ_IM;
  float* dftx_re = ws + OFF_DFTX_RE; float* dftx_im = ws + OFF_DFTX_IM;
  float* iwx_re  = ws + OFF_IWX_RE;  float* iwx_im  = ws + OFF_IWX_IM;
  float* cz_re   = ws + OFF_CZ_RE;   float* cz_im   = ws + OFF_CZ_IM;
  float* wyt_re  = ws + OFF_WYT_RE;  float* wyt_im  = ws + OFF_WYT_IM;
  float* iwy_re  = ws + OFF_IWY_RE;  float* iwy_im  = ws + OFF_IWY_IM;
  float* wint    = ws + OFF_WINT;    float* mwt     = ws + OFF_MWT;
  float* h   = ws + OFF_H;
  float* zre = ws + OFF_ZR, *zim = ws + OFF_ZI;
  float* yre = ws + OFF_YR, *yim = ws + OFF_YI;
  float* xre = ws + OFF_XR, *xim = ws + OFF_XI;
  float* ere = ws + OFF_ER, *eim = ws + OFF_EI;
  float* gre = ws + OFF_GR, *gim = ws + OFF_GI;
  float* hre = ws + OFF_HR, *him = ws + OFF_HI;
  float* m0  = ws + OFF_M0, *m1  = ws + OFF_M1;

  // --- constant DFT matrices + weight transposes (regenerated every call) ---
  fill_dft<<<dim3((128*16+255)/256), dim3(256), 0, stream>>>(dftz_re, dftz_im, 128, 16, 128, -1.f, 1.f, 0);
  fill_dft<<<dim3((16*128+255)/256), dim3(256), 0, stream>>>(dftx_re, dftx_im, 16, 128, 128, -1.f, 1.f, 0);
  fill_dft<<<dim3((128*16+255)/256), dim3(256), 0, stream>>>(iwx_re, iwx_im, 128, 16, 128, +1.f, 1.f/128.f, 0);
  fill_dft<<<dim3((16*128+255)/256), dim3(256), 0, stream>>>(cz_re, cz_im, 16, 128, 128, -1.f, 1.f/128.f, 1);
  transpose2d<<<dim3(1), dim3(256), 0, stream>>>(w_in, wint, 4, 64);
  for (int i = 0; i < 4; ++i)
    transpose2d<<<dim3((4096+255)/256), dim3(256), 0, stream>>>(mw[i], mwt + (size_t)i*4096, 64, 64);

  // --- input projection: h(64, 2^21) = w_in^T(64x4) @ x(4, 2^21) + b_in ---
  const int S0 = 128*128*128;
  launch_gemm(stream, wint, x, nullptr, nullptr, b_in, h,
              64, S0, 4, 4, S0, S0, 0, 0, 0, 1, 0.f, 1, 0);

  // --- 4 Fourier layers ---
  const int Ys[5] = {128, 65, 33, 17, 9};
  for (int L = 0; L < 4; ++L) {
    const int Y = Ys[L], ny2 = Ys[L+1];
    fill_dft<<<dim3((16*Y+255)/256), dim3(256), 0, stream>>>(wyt_re, wyt_im, 16, Y, Y, -1.f, 1.f, 0);
    fill_dft<<<dim3((ny2*16+255)/256), dim3(256), 0, stream>>>(iwy_re, iwy_im, ny2, 16, ny2, +1.f, 1.f/(float)ny2, 0);

    // (a) forward z: rows (c,x,y), K=128 -> 16 complex modes (planar)
    const int Mz = 64*128*Y;
    launch_gemm(stream, h, dftz_re, nullptr, nullptr, nullptr, zre,
                Mz, 16, 128, 128, 16, 16, 0, 0, 0, 1, 0.f, 0, 0);
    launch_gemm(stream, h, dftz_im, nullptr, nullptr, nullptr, zim,
                Mz, 16, 128, 128, 16, 16, 0, 0, 0, 1, 0.f, 0, 0);
    // (b) forward y: batch (c,x)=8192; C(16ky x 16kz) = Wy^T(16xY) @ Z(Yx16)
    launch_gemm(stream, wyt_re, zre, wyt_im, zim, nullptr, yre,
                16, 16, Y, Y, 16, 16, 0, (long long)Y*16, 256, 8192, -1.f, 0, 0);
    launch_gemm(stream, wyt_re, zim, wyt_im, zre, nullptr, yim,
                16, 16, Y, Y, 16, 16, 0, (long long)Y*16, 256, 8192, +1.f, 0, 0);
    // (c) forward x: batch c=64; C(16kx x 256) = Wx^T(16x128) @ Yb(128x256)
    launch_gemm(stream, dftx_re, yre, dftx_im, yim, nullptr, xre,
                16, 256, 128, 128, 256, 256, 0, 32768, 4096, 64, -1.f, 0, 0);
    launch_gemm(stream, dftx_re, yim, dftx_im, yre, nullptr, xim,
                16, 256, 128, 128, 256, 256, 0, 32768, 4096, 64, +1.f, 0, 0);
    // (d) per-mode 64x64 channel mix
    mix_modes<<<dim3(4096), dim3(64), 0, stream>>>(fws[L], xre, xim, ere, eim);
    // (e) inverse ky->y: batch (o,kx)=1024; C(ny2 x 16kz) = IWy(ny2x16) @ E(16x16)
    launch_gemm(stream, iwy_re, ere, iwy_im, eim, nullptr, gre,
                ny2, 16, 16, 16, 16, 16, 0, 256, (long long)ny2*16, 1024, -1.f, 0, 0);
    launch_gemm(stream, iwy_re, eim, iwy_im, ere, nullptr, gim,
                ny2, 16, 16, 16, 16, 16, 0, 256, (long long)ny2*16, 1024, +1.f, 0, 0);
    // (f) inverse kx->x: batch o=64; C(128 x ny2*16) = IWx(128x16) @ G(16 x ny2*16)
    const int NW = ny2*16;
    launch_gemm(stream, iwx_re, gre, iwx_im, gim, nullptr, hre,
                128, NW, 16, 16, NW, NW, 0, (long long)16*NW, (long long)128*NW, 64, -1.f, 0, 0);
    launch_gemm(stream, iwx_re, gim, iwx_im, gre, nullptr, him,
                128, NW, 16, 16, NW, NW, 0, (long long)16*NW, (long long)128*NW, 64, +1.f, 0, 0);
    // (g) irfft z + GELU: h(Mg x 128) = Hre@Cr + Him@Ci  (in-place over old h)
    const int Mg = 64*128*ny2;
    launch_gemm(stream, hre, cz_re, him, cz_im, nullptr, h,
                Mg, 128, 16, 16, 128, 128, 0, 0, 0, 1, +1.f, 0, 1);
  }

  // --- MLP head on (64, S4) channel-first; out = w_out^T @ ... + b_out ---
  const int S4 = 128*9*128;
  launch_gemm(stream, mwt + 0*4096, h,  nullptr, nullptr, mb[0], m0,
              64, S4, 64, 64, S4, S4, 0, 0, 0, 1, 0.f, 1, 1);
  launch_gemm(stream, mwt + 1*4096, m0, nullptr, nullptr, mb[1], m1,
              64, S4, 64, 64, S4, S4, 0, 0, 0, 1, 0.f, 1, 1);
  launch_gemm(stream, mwt + 2*4096, m1, nullptr, nullptr, mb[2], m0,
              64, S4, 64, 64, S4, S4, 0, 0, 0, 1, 0.f, 1, 1);
  launch_gemm(stream, mwt + 3*4096, m0, nullptr, nullptr, mb[3], m1,
              64, S4, 64, 64, S4, S4, 0, 0, 0, 1, 0.f, 1, 1);
  launch_gemm(stream, w_out, m1, nullptr, nullptr, b_out, out,
              1, S4, 64, 64, S4, S4, 0, 0, 0, 1, 0.f, 1, 0);
}